// PolicyHeadWithAttention_16844861735073
// MI455X (gfx1250) — compile-verified
//
#include <hip/hip_runtime.h>
#include <math.h>

// Problem constants (from reference)
#define B_SZ   4096
#define C_INCH 256
#define S_SZ   90          // board cells = seq len
#define KFC    180         // E*S = fc input dim
#define AOUT   2086        // fc output dim
#define BN_EPS 1e-5f
#define NTILES ((AOUT + 15) / 16)   // 131

typedef float v2f __attribute__((ext_vector_type(2)));
typedef float v8f __attribute__((ext_vector_type(8)));

// ---------------------------------------------------------------------------
// Kernel 1: conv1x1(256->2) + BN + 2-head attention (head_dim=1) + out_proj.
// One block per batch element; memory-bound on streaming x[b] (90 KB).
// Writes flat[b][e*90 + s] (fp32) to workspace.
// ---------------------------------------------------------------------------
__global__ __launch_bounds__(96) void head_kernel(
    const float* __restrict__ x,        // [B,256,90]
    const float* __restrict__ conv_w,   // [2,256]
    const float* __restrict__ bn_gamma, const float* __restrict__ bn_beta,
    const float* __restrict__ bn_mean,  const float* __restrict__ bn_var,
    const float* __restrict__ ipw,      // [6,2] packed QKV weights
    const float* __restrict__ ipb,      // [6]
    const float* __restrict__ opw,      // [2,2]
    const float* __restrict__ opb,      // [2]
    float* __restrict__ flat)           // [B,180]
{
  __shared__ float w0[C_INCH];
  __shared__ float w1[C_INCH];
  __shared__ float kh[2][S_SZ];
  __shared__ float vh[2][S_SZ];

  const int b = blockIdx.x;
  const int t = threadIdx.x;

  for (int c = t; c < C_INCH; c += 96) {
    w0[c] = conv_w[c];
    w1[c] = conv_w[C_INCH + c];
  }
  __syncthreads();

  float q0 = 0.f, q1 = 0.f;
  if (t < S_SZ) {
    // conv: dot(x[b,:,s], conv_w[e,:])  -- coalesced across s=lane
    const float* xb = x + ((size_t)b * C_INCH * S_SZ) + t;
    float a0 = 0.f, a1 = 0.f;
#pragma unroll 8
    for (int c = 0; c < C_INCH; ++c) {
      float xv = xb[(size_t)c * S_SZ];
      a0 = fmaf(xv, w0[c], a0);
      a1 = fmaf(xv, w1[c], a1);
    }
    // BatchNorm (inference)
    float sc0 = bn_gamma[0] * rsqrtf(bn_var[0] + BN_EPS);
    float sc1 = bn_gamma[1] * rsqrtf(bn_var[1] + BN_EPS);
    float y0 = fmaf(a0, sc0, bn_beta[0] - bn_mean[0] * sc0);
    float y1 = fmaf(a1, sc1, bn_beta[1] - bn_mean[1] * sc1);
    // packed QKV projection; head h uses row h (Q), 2+h (K), 4+h (V)
    q0        = ipw[0]  * y0 + ipw[1]  * y1 + ipb[0];
    q1        = ipw[2]  * y0 + ipw[3]  * y1 + ipb[1];
    kh[0][t]  = ipw[4]  * y0 + ipw[5]  * y1 + ipb[2];
    kh[1][t]  = ipw[6]  * y0 + ipw[7]  * y1 + ipb[3];
    vh[0][t]  = ipw[8]  * y0 + ipw[9]  * y1 + ipb[4];
    vh[1][t]  = ipw[10] * y0 + ipw[11] * y1 + ipb[5];
  }
  __syncthreads();

  if (t < S_SZ) {
    float o[2];
#pragma unroll
    for (int h = 0; h < 2; ++h) {
      const float qi = (h == 0) ? q0 : q1;   // head_dim=1, scale = 1/sqrt(1)
      float mx = -3.402823466e38f;
      for (int j = 0; j < S_SZ; ++j) mx = fmaxf(mx, qi * kh[h][j]);
      float se = 0.f, sv = 0.f;
      for (int j = 0; j < S_SZ; ++j) {
        float e = __expf(qi * kh[h][j] - mx);
        se += e;
        sv = fmaf(e, vh[h][j], sv);
      }
      o[h] = sv / se;
    }
    float p0 = opw[0] * o[0] + opw[1] * o[1] + opb[0];
    float p1 = opw[2] * o[0] + opw[3] * o[1] + opb[1];
    flat[(size_t)b * KFC + t]        = p0;   // channel 0 (e*90 + s)
    flat[(size_t)b * KFC + S_SZ + t] = p1;   // channel 1
  }
}

// ---------------------------------------------------------------------------
// Kernel 2: logits = flat[4096,180] @ fc_w^T[180,2086] + fc_b via
// V_WMMA_F32_16X16X4_F32. Block: 32-row M-slab (two 16-row A tiles in LDS),
// 8 waves each owning a 16-col N-tile. Each wave keeps TWO independent
// accumulator tiles so every B load (shared) feeds two WMMAs that pipeline
// without same-D RAW serialization, and fc_w L2 traffic is halved.
// ---------------------------------------------------------------------------
#define ASTRIDE 182   // LDS row stride in floats (8B aligned, spreads banks)
#define MROWS   32    // rows per block (2 WMMA M-tiles)

__global__ __launch_bounds__(256) void fc_kernel(
    const float* __restrict__ flat,   // [4096,180]
    const float* __restrict__ fcw,    // [2086,180] row-major (out,in)
    const float* __restrict__ fcb,    // [2086]
    float* __restrict__ out)          // [4096,2086]
{
  __shared__ float As[MROWS * ASTRIDE];

  const int m0  = blockIdx.x * MROWS;
  const int tid = threadIdx.x;

  // Stage A slab: 32 rows x 180 cols
  for (int idx = tid; idx < MROWS * KFC; idx += 256) {
    int r = idx / KFC;
    int c = idx - r * KFC;
    As[r * ASTRIDE + c] = flat[(size_t)(m0 + r) * KFC + c];
  }
  __syncthreads();

  const int wave = tid >> 5;
  const int lane = tid & 31;
  const int nt   = blockIdx.y * 8 + wave;
  if (nt >= NTILES) return;

  const int n0    = nt * 16;
  const int n     = lane & 15;
  const int col   = n0 + n;
  const int ncl   = (col < AOUT) ? col : (AOUT - 1);  // clamp edge reads
  const int khalf = (lane >> 4) << 1;                 // 0 (lanes 0-15) or 2

  // A operand (16x4 f32): lanes 0-15 hold K=0,1 ; lanes 16-31 hold K=2,3
  const float* ap0 = &As[(lane & 15) * ASTRIDE + khalf];
  const float* ap1 = ap0 + 16 * ASTRIDE;
  // B operand (4x16 f32): B[k][n] = fc_w[n][k]; rows contiguous in K -> b64
  const float* bp = fcw + (size_t)ncl * KFC + khalf;

  v8f acc0 = {0.f, 0.f, 0.f, 0.f, 0.f, 0.f, 0.f, 0.f};
  v8f acc1 = {0.f, 0.f, 0.f, 0.f, 0.f, 0.f, 0.f, 0.f};

#pragma unroll 5
  for (int t = 0; t < KFC / 4; ++t) {
    v2f a0 = *(const v2f*)(ap0 + 4 * t);
    v2f a1 = *(const v2f*)(ap1 + 4 * t);
    v2f bb = *(const v2f*)(bp  + 4 * t);
    __builtin_prefetch(bp + 4 * (t + 16), 0, 3);  // near-scope prefetch ahead
    acc0 = __builtin_amdgcn_wmma_f32_16x16x4_f32(
        false, a0, false, bb, (short)0, acc0, false, false);
    acc1 = __builtin_amdgcn_wmma_f32_16x16x4_f32(
        false, a1, false, bb, (short)0, acc1, false, false);
  }

  if (col < AOUT) {
    const float bias = fcb[col];
    // D layout: VGPR r -> row base+r (lanes 0-15) or base+8+r (lanes 16-31)
    const int rbase = (lane >> 4) << 3;
    float* orow0 = out + (size_t)(m0 + rbase) * AOUT + col;
    float* orow1 = out + (size_t)(m0 + 16 + rbase) * AOUT + col;
#pragma unroll
    for (int r = 0; r < 8; ++r) {
      orow0[(size_t)r * AOUT] = acc0[r] + bias;
      orow1[(size_t)r * AOUT] = acc1[r] + bias;
    }
  }
}

// ---------------------------------------------------------------------------
extern "C" void kernel_launch(void* const* d_in, const int* in_sizes, int n_in,
                              void* d_out, int out_size, void* d_ws, size_t ws_size,
                              hipStream_t stream) {
  const float* x        = (const float*)d_in[0];
  const float* conv_w   = (const float*)d_in[1];
  const float* bn_gamma = (const float*)d_in[2];
  const float* bn_beta  = (const float*)d_in[3];
  const float* bn_mean  = (const float*)d_in[4];
  const float* bn_var   = (const float*)d_in[5];
  const float* ipw      = (const float*)d_in[6];
  const float* ipb      = (const float*)d_in[7];
  const float* opw      = (const float*)d_in[8];
  const float* opb      = (const float*)d_in[9];
  const float* fcw      = (const float*)d_in[10];
  const float* fcb      = (const float*)d_in[11];

  float* flat = (float*)d_ws;               // [4096,180] fp32 = 2.95 MB
  float* outp = (float*)d_out;              // [4096,2086] fp32

  head_kernel<<<B_SZ, 96, 0, stream>>>(x, conv_w, bn_gamma, bn_beta, bn_mean,
                                       bn_var, ipw, ipb, opw, opb, flat);

  dim3 grid2(B_SZ / MROWS, (NTILES + 7) / 8);  // 128 x 17
  fc_kernel<<<grid2, 256, 0, stream>>>(flat, fcw, fcb, outp);
}